// HACRF_57535381897353
// MI455X (gfx1250) — compile-verified
//
#include <hip/hip_runtime.h>
#include <hip/hip_bf16.h>

// CRF forward/labeled scores for MI455X (gfx1250, wave32).
// B=1024, S=1024, L=5, START=3, END=4.

#define CRF_B 1024
#define CRF_S 1024
#define CRF_L 5
#define CRF_START 3
#define CRF_END 4
#define TT 16               // time steps per staged chunk
#define ROWP 85             // padded LDS row stride in DWORDs (80 + 80/16 pads)
#define SCAN_WAVES 4        // waves per block in scan kernel
#define LOG2E 1.4426950408889634f
#define LN2 0.6931471805599453f

typedef __attribute__((ext_vector_type(2))) float v2f;
typedef __attribute__((ext_vector_type(8))) float v8f;
typedef __attribute__((ext_vector_type(4))) unsigned int u32x4;
typedef __attribute__((ext_vector_type(4))) int i32x4;
typedef __attribute__((ext_vector_type(8))) int i32x8;

#if defined(__has_builtin)
#if __has_builtin(__builtin_amdgcn_tensor_load_to_lds)
#define HAVE_TDM 1
#endif
#endif
#ifndef HAVE_TDM
#define HAVE_TDM 0
#endif

#if HAVE_TDM
// Issue one TDM descriptor: 2D tile, 16 rows (batch) x 80 floats (16 steps * 5
// labels, contiguous), global row stride S*L floats, LDS padded 1 DWORD / 16.
// This toolchain exposes the 6-argument builtin form:
//   (uint32x4 g0, int32x8 g1, int32x4 g2, int32x4 g3, int32x8 g4, i32 cpol)
__device__ __forceinline__ void tdm_load_chunk(const float* gptr,
                                               unsigned lds_byte_addr) {
  unsigned long long ga = (unsigned long long)(uintptr_t)gptr;
  u32x4 g0;
  g0[0] = 1u;                                   // count=1, user descriptor
  g0[1] = lds_byte_addr;                        // LDS dest (bytes)
  g0[2] = (unsigned)ga;                         // global_addr[31:0]
  g0[3] = (unsigned)((ga >> 32) & 0x01FFFFFFu)  // global_addr[56:32]
          | (2u << 30);                         // type = 2 ("image")
  i32x8 g1;
  g1[0] = (int)((2u << 16)      // data_size = 4B
                | (1u << 20)    // pad_enable
                | (3u << 22)    // pad_interval = 16 DWORDs
                | (0u << 25));  // pad_amount = 1 DWORD
  g1[1] = (int)(80u << 16);     // tensor_dim0 = 80 (bits 79:48, low half)
  g1[2] = (int)(16u << 16);     // tensor_dim1 = 16 (bits 95:80)
  g1[3] = (int)(80u << 16);     // tile_dim0 = 80 (bits 127:112)
  g1[4] = 16;                   // tile_dim1 = 16 (bits 143:128)
  g1[5] = CRF_S * CRF_L;        // tensor_dim0_stride = 5120 elements
  g1[6] = 0;
  g1[7] = 0;
  i32x4 z4 = {0, 0, 0, 0};      // groups 2/3 unused (2D tensor)
  i32x8 z8 = {0, 0, 0, 0, 0, 0, 0, 0};
  __builtin_amdgcn_tensor_load_to_lds(g0, g1, z4, z4, z8, 0);
}
#endif

// ---------------------------------------------------------------------------
// Kernel 1: exp-domain forward scan, 16 batch rows per wave via WMMA f32
// 16x16x4. gamma' = E^T * gamma (E = exp(T), zero-padded), then
// gamma' *= exp(x[b,t,:]); renormalize by per-batch max every 4 steps
// (per-lane only). Scale bookkeeping kept in log2 domain with raw
// v_rcp_f32 / v_log_f32 / v_exp_f32 so nothing on the sequential chain
// expands into IEEE fixup sequences.
// ---------------------------------------------------------------------------
__global__ __launch_bounds__(32 * SCAN_WAVES) void crf_scan_kernel(
    const float* __restrict__ x,      // [B,S,L]
    const int* __restrict__ lens,     // [B]
    const float* __restrict__ T,      // [L,L]
    float* __restrict__ unlab_part) { // [B]
  __shared__ float xs[SCAN_WAVES][2][16 * ROWP];  // double-buffered raw x tiles

  const int lane = threadIdx.x & 31;
  const int w    = threadIdx.x >> 5;
  const int wid  = blockIdx.x * SCAN_WAVES + w;
  const int b0   = wid * 16;
  const int m    = lane & 15;       // batch within tile / column of D
  const bool up  = lane >= 16;
  const bool jv  = (m < CRF_L);

  // A = E^T in WMMA A-layout: lanes 0-15: M=j, K=0 (.x), K=1 (.y);
  //                           lanes16-31: M=j, K=2 (.x), K=3 (.y).
  v2f A1, A2;
  A1.x = jv ? __expf(T[(up ? 2 : 0) * CRF_L + m]) : 0.f;
  A1.y = jv ? __expf(T[(up ? 3 : 1) * CRF_L + m]) : 0.f;
  A2.x = (jv && !up) ? __expf(T[4 * CRF_L + m]) : 0.f;
  A2.y = 0.f;

  const float e20 = __expf(T[0 * CRF_L + CRF_END]);
  const float e21 = __expf(T[1 * CRF_L + CRF_END]);
  const float e22 = __expf(T[2 * CRF_L + CRF_END]);
  const float e23 = __expf(T[3 * CRF_L + CRF_END]);
  const float e24 = __expf(T[4 * CRF_L + CRF_END]);

  const float* xb   = x + (size_t)(b0 + m) * CRF_S * CRF_L;
  const int lenm1   = lens[b0 + m] - 1;

  // t = 0 init.
  float g0 = up ? 0.f : __expf(T[CRF_START * CRF_L + 0] + xb[0]);
  float g1 = up ? 0.f : __expf(T[CRF_START * CRF_L + 1] + xb[1]);
  float g2 = up ? 0.f : __expf(T[CRF_START * CRF_L + 2] + xb[2]);
  float g3 = up ? 0.f : __expf(T[CRF_START * CRF_L + 3] + xb[3]);
  float g4 = up ? 0.f : __expf(T[CRF_START * CRF_L + 4] + xb[4]);
  float ls2 = 0.f;  // log2 of accumulated scale

  bool cap = (lenm1 == 0);
  float s0 = cap ? g0 : 0.f, s1 = cap ? g1 : 0.f, s2 = cap ? g2 : 0.f;
  float s3 = cap ? g3 : 0.f, s4 = cap ? g4 : 0.f;
  float savedls2 = 0.f;

#if HAVE_TDM
  const unsigned lds0 = (unsigned)(uintptr_t)&xs[w][0][0];
  const unsigned lds1 = (unsigned)(uintptr_t)&xs[w][1][0];
  tdm_load_chunk(x + (size_t)b0 * CRF_S * CRF_L, lds0);
#endif

  for (int tc = 0; tc < CRF_S; tc += TT) {
    const int cur   = (tc / TT) & 1;
    const bool more = (tc + TT < CRF_S);
    float* bb = &xs[w][cur][0];

#if HAVE_TDM
    if (more) {
      tdm_load_chunk(x + ((size_t)b0 * CRF_S + tc + TT) * CRF_L,
                     cur ? lds0 : lds1);
      __builtin_amdgcn_s_wait_tensorcnt(1);
    } else {
      __builtin_amdgcn_s_wait_tensorcnt(0);
    }
    asm volatile("" ::: "memory");  // keep ds_loads after the wait
#else
    if (more) __builtin_prefetch(x + ((size_t)(b0 + m) * CRF_S + tc + TT) * CRF_L, 0, 0);
    for (int k = lane; k < 16 * TT * CRF_L; k += 32) {
      int bi = k / (TT * CRF_L);
      int kk = k - bi * (TT * CRF_L);
      bb[bi * ROWP + kk + (kk >> 4)] =
          x[((size_t)(b0 + bi) * CRF_S + tc) * CRF_L + kk];
    }
#endif
    const float* lanebuf = bb + m * ROWP;

    for (int tt = (tc == 0 ? 1 : 0); tt < TT; ++tt) {
      const int t = tc + tt;

      // B = gamma^T in WMMA B-layout: V0 = {K=0 | K=2}, V1 = {K=1 | K=3}.
      float h2 = __shfl(g2, m, 32);
      float h3 = __shfl(g3, m, 32);
      v2f B1, B2;
      B1.x = up ? h2 : g0;
      B1.y = up ? h3 : g1;
      B2.x = up ? 0.f : g4;
      B2.y = 0.f;

      v8f c = {0.f, 0.f, 0.f, 0.f, 0.f, 0.f, 0.f, 0.f};
      c = __builtin_amdgcn_wmma_f32_16x16x4_f32(false, A1, false, B1,
                                                (short)0, c, false, false);
      c = __builtin_amdgcn_wmma_f32_16x16x4_f32(false, A2, false, B2,
                                                (short)0, c, false, false);

      // Factors exp(x[b,t,j]) from the padded LDS tile (raw v_exp_f32).
      const int k0 = tt * CRF_L;
      float f0 = __builtin_amdgcn_exp2f(lanebuf[k0 + (k0 >> 4)] * LOG2E);
      const int k1 = k0 + 1;
      float f1 = __builtin_amdgcn_exp2f(lanebuf[k1 + (k1 >> 4)] * LOG2E);
      const int k2 = k0 + 2;
      float f2 = __builtin_amdgcn_exp2f(lanebuf[k2 + (k2 >> 4)] * LOG2E);
      const int k3 = k0 + 3;
      float f3 = __builtin_amdgcn_exp2f(lanebuf[k3 + (k3 >> 4)] * LOG2E);
      const int k4 = k0 + 4;
      float f4 = __builtin_amdgcn_exp2f(lanebuf[k4 + (k4 >> 4)] * LOG2E);

      g0 = c[0] * f0;
      g1 = c[1] * f1;
      g2 = c[2] * f2;
      g3 = c[3] * f3;
      g4 = c[4] * f4;

      // Renormalize every 4 steps (growth <= ~2^61 between renorms).
      if ((tt & 3) == 3) {
        float mx  = fmaxf(fmaxf(fmaxf(g0, g1), fmaxf(g2, g3)),
                          fmaxf(g4, 1e-30f));
        float inv = __builtin_amdgcn_rcpf(mx);
        g0 *= inv; g1 *= inv; g2 *= inv; g3 *= inv; g4 *= inv;
        ls2 -= __builtin_amdgcn_logf(inv);  // exact bookkeeping of applied scale
      }

      // Capture alpha at t = len-1 (exact regardless of renorm cadence).
      bool c2 = (t == lenm1);
      s0 = c2 ? g0 : s0;  s1 = c2 ? g1 : s1;  s2 = c2 ? g2 : s2;
      s3 = c2 ? g3 : s3;  s4 = c2 ? g4 : s4;
      savedls2 = c2 ? ls2 : savedls2;
    }
  }

  // Readout: normalize saved vector once (it may be up to ~2^61), then
  // lse_j(alpha_last[j] + T[j,END]) = ln2*(log2(sum) - log2(invs) + savedls2).
  float mxs  = fmaxf(fmaxf(fmaxf(s0, s1), fmaxf(s2, s3)), fmaxf(s4, 1e-30f));
  float invs = __builtin_amdgcn_rcpf(mxs);
  float sum  = (s0 * invs) * e20 + (s1 * invs) * e21 + (s2 * invs) * e22 +
               (s3 * invs) * e23 + (s4 * invs) * e24;
  float res  = (__builtin_amdgcn_logf(sum) - __builtin_amdgcn_logf(invs) +
                savedls2) * LN2;
  if (!up) unlab_part[b0 + m] = res;
}

// ---------------------------------------------------------------------------
// Kernel 2: labeled path score, one wave per batch row, coalesced over t.
// ---------------------------------------------------------------------------
__global__ __launch_bounds__(256) void crf_labeled_kernel(
    const float* __restrict__ x, const int* __restrict__ lens,
    const int* __restrict__ tags, const float* __restrict__ T,
    float* __restrict__ lab_part) {
  const int lane = threadIdx.x & 31;
  const int w    = threadIdx.x >> 5;
  const int b    = blockIdx.x * 8 + w;
  const int len  = lens[b];
  const int* tg  = tags + (size_t)b * CRF_S;
  const float* xb = x + (size_t)b * CRF_S * CRF_L;

  float part = 0.f;
  for (int t = lane; t < CRF_S; t += 32) {
    int tag = tg[t];
    if (t == 0) {
      part += T[CRF_START * CRF_L + tag] + xb[tag];          // begin
    } else if (t < len) {
      int tp = tg[t - 1];
      part += T[tp * CRF_L + tag] + xb[t * CRF_L + tag];     // masked middle
    }
    if (t == len - 1) part += T[tag * CRF_L + CRF_END];      // end
  }
  for (int off = 16; off; off >>= 1) part += __shfl_xor(part, off, 32);
  if (lane == 0) lab_part[b] = part;
}

// ---------------------------------------------------------------------------
// Kernel 3: deterministic fixed-order final reductions -> d_out[0], d_out[1].
// ---------------------------------------------------------------------------
__global__ __launch_bounds__(256) void crf_reduce_kernel(
    const float* __restrict__ un, const float* __restrict__ lab,
    float* __restrict__ out) {
  __shared__ float su[256];
  __shared__ float sl[256];
  const int tid = threadIdx.x;
  float a = 0.f, b = 0.f;
  for (int k = tid; k < CRF_B; k += 256) { a += un[k]; b += lab[k]; }
  su[tid] = a; sl[tid] = b;
  __syncthreads();
  for (int s = 128; s; s >>= 1) {
    if (tid < s) { su[tid] += su[tid + s]; sl[tid] += sl[tid + s]; }
    __syncthreads();
  }
  if (tid == 0) { out[0] = su[0]; out[1] = sl[0]; }
}

extern "C" void kernel_launch(void* const* d_in, const int* in_sizes, int n_in,
                              void* d_out, int out_size, void* d_ws, size_t ws_size,
                              hipStream_t stream) {
  const float* x    = (const float*)d_in[0];  // lstm_scores [B,S,L]
  const int*   lens = (const int*)d_in[1];    // word_seq_lens [B]
  const int*   tags = (const int*)d_in[2];    // tags [B,S]
  // d_in[3] = mask (recomputed from lens instead; avoids bool-dtype ambiguity)
  const float* T    = (const float*)d_in[4];  // transition [L,L]

  float* ws_un  = (float*)d_ws;
  float* ws_lab = ws_un + CRF_B;

  crf_scan_kernel<<<CRF_B / 16 / SCAN_WAVES, 32 * SCAN_WAVES, 0, stream>>>(
      x, lens, T, ws_un);
  crf_labeled_kernel<<<CRF_B / 8, 256, 0, stream>>>(x, lens, tags, T, ws_lab);
  crf_reduce_kernel<<<1, 256, 0, stream>>>(ws_un, ws_lab, (float*)d_out);
}